// ChamferLoss_40810779247121
// MI455X (gfx1250) — compile-verified
//
#include <hip/hip_runtime.h>

typedef __attribute__((ext_vector_type(2))) float v2f;
typedef __attribute__((ext_vector_type(8))) float v8f;

// Problem constants from the reference: B=16, N=M=2048, D=3 (pad K to 4).
#define CH_B 16
#define CH_N 2048
#define CH_M 2048
#define WAVES_PER_BLOCK 4   // 128 threads = 4 wave32s

__global__ void chamfer_zero_kernel(float* out) { out[0] = 0.0f; }

// One direction of Chamfer: for each point of P (rows), min squared distance
// to any point of Q (cols); atomically accumulates sum(min)/count into *out.
// d(r,c) = |p_r|^2 + |q_c|^2 - 2 p_r . q_c, with the -2 p.q term computed by
// V_WMMA_F32_16X16X4_F32 (A pre-scaled by -2, K=3 padded to 4 with zeros).
// |p_r|^2 is constant along the min axis, so it is added AFTER the column min
// (halves the per-tile fixup VALU work in the hot loop).
__global__ void __launch_bounds__(32 * WAVES_PER_BLOCK)
chamfer_dir_kernel(const float* __restrict__ P,  // [CH_B, CH_N, 3] query rows
                   const float* __restrict__ Q,  // [CH_B, CH_M, 3] target cols
                   float* __restrict__ out,
                   float inv_count) {
  const int wave = threadIdx.x >> 5;
  const int lane = threadIdx.x & 31;
  const int tile = blockIdx.x * WAVES_PER_BLOCK + wave;  // global 16-row tile id
  const int bidx = tile >> 7;            // tile / (CH_N/16) ; CH_N/16 == 128
  const int row0 = (tile & 127) << 4;    // 16-row tile base within the batch

  const float* pb = P + (size_t)bidx * CH_N * 3;
  const float* qb = Q + (size_t)bidx * CH_M * 3;

  const int  r  = lane & 15;
  const bool hi = lane >= 16;

  // Load this wave's 16 query points (each half-wave reads the same 16 rows).
  const float px = pb[(row0 + r) * 3 + 0];
  const float py = pb[(row0 + r) * 3 + 1];
  const float pz = pb[(row0 + r) * 3 + 2];
  const float a2 = px * px + py * py + pz * pz;

  // A matrix (16x4 f32), scaled by -2:
  //   lanes 0-15 : VGPR0 = K0 (x), VGPR1 = K1 (y)   for row M = lane
  //   lanes 16-31: VGPR0 = K2 (z), VGPR1 = K3 (0)   for row M = lane-16
  v2f a;
  a.x = hi ? (-2.0f * pz) : (-2.0f * px);
  a.y = hi ? 0.0f         : (-2.0f * py);

  // |p|^2 of the 8 C/D rows this lane's accumulators correspond to
  // (row = v + 8*hi); only needed after the column-min, in the tail.
  float arow[8];
#pragma unroll
  for (int v = 0; v < 8; ++v)
    arow[v] = __shfl(a2, v + (hi ? 8 : 0), 32);

  float rowmin[8];
#pragma unroll
  for (int v = 0; v < 8; ++v) rowmin[v] = 3.4e38f;

  for (int col0 = 0; col0 < CH_M; col0 += 16) {
    // B matrix (4x16 f32): column N = lane&15.
    //   lanes 0-15 : VGPR0 = K0 (x), VGPR1 = K1 (y)
    //   lanes 16-31: VGPR0 = K2 (z), VGPR1 = K3 (0)
    const float gx = qb[(col0 + r) * 3 + 0];
    const float gy = qb[(col0 + r) * 3 + 1];
    const float gz = qb[(col0 + r) * 3 + 2];
    const float b2 = gx * gx + gy * gy + gz * gz;
    v2f bm;
    bm.x = hi ? gz   : gx;
    bm.y = hi ? 0.0f : gy;

    v8f c = {};
    // 8 args: (neg_a, A, neg_b, B, c_mod, C, reuse_a, reuse_b)
    c = __builtin_amdgcn_wmma_f32_16x16x4_f32(
        false, a, false, bm, (short)0, c, false, false);

    // Partial distance (row |p|^2 deferred): c[v] + |q_col|^2
#pragma unroll
    for (int v = 0; v < 8; ++v)
      rowmin[v] = fminf(rowmin[v], c[v] + b2);
  }

  // Min-reduce each row across its 16 covering lanes (xor offsets <=8 stay
  // within the 0-15 / 16-31 halves on wave32).
#pragma unroll
  for (int v = 0; v < 8; ++v) {
    float m = rowmin[v];
    m = fminf(m, __shfl_xor(m, 8, 32));
    m = fminf(m, __shfl_xor(m, 4, 32));
    m = fminf(m, __shfl_xor(m, 2, 32));
    m = fminf(m, __shfl_xor(m, 1, 32));
    rowmin[v] = m;
  }

  // Lane 0 holds rows row0+0..7, lane 16 holds rows row0+8..15; apply the
  // deferred |p_row|^2 term here (arow is uniform within each 16-lane half).
  if (r == 0) {
    float s = 0.0f;
#pragma unroll
    for (int v = 0; v < 8; ++v) s += rowmin[v] + arow[v];
    atomicAdd(out, s * inv_count);
  }
}

extern "C" void kernel_launch(void* const* d_in, const int* in_sizes, int n_in,
                              void* d_out, int out_size, void* d_ws, size_t ws_size,
                              hipStream_t stream) {
  const float* pred = (const float*)d_in[0];  // [16, 2048, 3] f32
  const float* gt   = (const float*)d_in[1];  // [16, 2048, 3] f32
  float* out = (float*)d_out;                 // scalar f32 loss

  chamfer_zero_kernel<<<dim3(1), dim3(1), 0, stream>>>(out);

  const int row_tiles = CH_B * (CH_N / 16);          // 2048 sixteen-row tiles
  dim3 grid(row_tiles / WAVES_PER_BLOCK);            // 512 blocks
  dim3 block(32 * WAVES_PER_BLOCK);                  // 128 threads (4 waves)

  // dist1: for each pred point, nearest gt point; mean over B*N.
  chamfer_dir_kernel<<<grid, block, 0, stream>>>(
      pred, gt, out, 1.0f / (float)(CH_B * CH_N));
  // dist2: for each gt point, nearest pred point; mean over B*M.
  chamfer_dir_kernel<<<grid, block, 0, stream>>>(
      gt, pred, out, 1.0f / (float)(CH_B * CH_M));
}